// AgenticDTT_38817914421522
// MI455X (gfx1250) — compile-verified
//
#include <hip/hip_runtime.h>
#include <hip/hip_bf16.h>
#include <math.h>

#define Bn 16
#define Sn 512
#define STATEn 128
#define ACTn 32
#define En 512
#define Hn 8
#define DHn 64
#define Ln 6
#define NINn 7
#define NLEAFn 8
#define FHn 512
#define Mrows (Bn * Sn)   // 8192

#define FA_ROWS 64        // q rows per workgroup in fused attention
#define FA_STRIDE 516     // LDS row stride (floats): 516%64==4 -> no bank conflicts
#define FA_LDS_BYTES (FA_ROWS * FA_STRIDE * 4)  // 132 KB of 320 KB/WGP

typedef float v2f __attribute__((ext_vector_type(2)));
typedef float v8f __attribute__((ext_vector_type(8)));

static __device__ __forceinline__ v8f wmma4(v2f a, v2f b, v8f c) {
  return __builtin_amdgcn_wmma_f32_16x16x4_f32(false, a, false, b, (short)0, c,
                                               false, false);
}

// ---------------------------------------------------------------------------
// FAST batched GEMM on the CDNA5 matrix core: C[z] = A[z] @ B[z] (+bias[z])
// Requirements: M%128==0, N%32==0, K%4==0. One wave owns a 32x32 tile.
// ---------------------------------------------------------------------------
template <bool TRANSB, int ACT, bool ACCUM>
__global__ __launch_bounds__(128) void gemm_wmma_fast(
    const float* __restrict__ A, int lda, long sA,
    const float* __restrict__ W, int ldw, long sW,
    const float* __restrict__ bias, long sBias,
    float* __restrict__ C, int ldc, long sC, int K) {
  const int z = blockIdx.z;
  A += (long)z * sA;
  W += (long)z * sW;
  C += (long)z * sC;
  if (bias) bias += (long)z * sBias;
  const int lane = threadIdx.x;
  const int half = lane >> 4;  // 0 or 1
  const int l16 = lane & 15;
  const int tm = (blockIdx.y * 4 + threadIdx.y) * 32;
  const int tn = blockIdx.x * 32;
  const int kh = half * 2;

  const float* Arow0 = A + (long)(tm + l16) * lda + kh;
  const float* Arow1 = A + (long)(tm + 16 + l16) * lda + kh;
  v8f acc00 = {}, acc01 = {}, acc10 = {}, acc11 = {};

  if (TRANSB) {
    const float* Brow0 = W + (long)(tn + l16) * ldw + kh;
    const float* Brow1 = W + (long)(tn + 16 + l16) * ldw + kh;
#pragma unroll 4
    for (int k0 = 0; k0 < K; k0 += 4) {
      const v2f a0 = *(const v2f*)(Arow0 + k0);
      const v2f a1 = *(const v2f*)(Arow1 + k0);
      const v2f b0 = *(const v2f*)(Brow0 + k0);
      const v2f b1 = *(const v2f*)(Brow1 + k0);
      acc00 = wmma4(a0, b0, acc00);
      acc01 = wmma4(a0, b1, acc01);
      acc10 = wmma4(a1, b0, acc10);
      acc11 = wmma4(a1, b1, acc11);
    }
  } else {
    const float* Wc0 = W + tn + l16;
    const float* Wc1 = W + tn + 16 + l16;
#pragma unroll 4
    for (int k0 = 0; k0 < K; k0 += 4) {
      const v2f a0 = *(const v2f*)(Arow0 + k0);
      const v2f a1 = *(const v2f*)(Arow1 + k0);
      const long r0 = (long)(k0 + kh) * ldw;
      const long r1 = r0 + ldw;
      v2f b0, b1;
      b0.x = Wc0[r0];
      b0.y = Wc0[r1];
      b1.x = Wc1[r0];
      b1.y = Wc1[r1];
      acc00 = wmma4(a0, b0, acc00);
      acc01 = wmma4(a0, b1, acc01);
      acc10 = wmma4(a1, b0, acc10);
      acc11 = wmma4(a1, b1, acc11);
    }
  }

  const int c0 = tn + l16;
  const int c1 = tn + 16 + l16;
  const float bv0 = bias ? bias[c0] : 0.f;
  const float bv1 = bias ? bias[c1] : 0.f;
  const int rb = half * 8;
#pragma unroll
  for (int r = 0; r < 8; ++r) {
    float v00 = acc00[r] + bv0;
    float v01 = acc01[r] + bv1;
    float v10 = acc10[r] + bv0;
    float v11 = acc11[r] + bv1;
    if (ACT == 1) {
      v00 = fmaxf(v00, 0.f); v01 = fmaxf(v01, 0.f);
      v10 = fmaxf(v10, 0.f); v11 = fmaxf(v11, 0.f);
    }
    if (ACT == 2) {
      v00 = tanhf(v00); v01 = tanhf(v01);
      v10 = tanhf(v10); v11 = tanhf(v11);
    }
    const long i0 = (long)(tm + rb + r) * ldc;
    const long i1 = (long)(tm + 16 + rb + r) * ldc;
    if (ACCUM) {
      C[i0 + c0] += v00; C[i0 + c1] += v01;
      C[i1 + c0] += v10; C[i1 + c1] += v11;
    } else {
      C[i0 + c0] = v00; C[i0 + c1] = v01;
      C[i1 + c0] = v10; C[i1 + c1] = v11;
    }
  }
}

// ---------------------------------------------------------------------------
// Guarded reference GEMM (only for tiny odd shapes: N=7, N=1, K=1)
// ---------------------------------------------------------------------------
template <bool TRANSB>
__global__ __launch_bounds__(128) void gemm_wmma_ref(
    const float* __restrict__ A, int lda, long sA,
    const float* __restrict__ W, int ldw, long sW,
    const float* __restrict__ bias,
    float* __restrict__ C, int ldc, long sC, int M, int N, int K) {
  const int z = blockIdx.z;
  A += (long)z * sA;
  W += (long)z * sW;
  C += (long)z * sC;
  const int lane = threadIdx.x;
  const int half = lane >> 4;
  const int l16 = lane & 15;
  const int tm = (blockIdx.y * 4 + threadIdx.y) * 16;
  const int tn = blockIdx.x * 16;
  const int am = tm + l16;
  const int bn = tn + l16;
  v8f acc = {};
  for (int k0 = 0; k0 < K; k0 += 4) {
    const int ka = k0 + half * 2;
    v2f a, b;
    a.x = (am < M && ka < K) ? A[(long)am * lda + ka] : 0.f;
    a.y = (am < M && ka + 1 < K) ? A[(long)am * lda + ka + 1] : 0.f;
    if (TRANSB) {
      b.x = (bn < N && ka < K) ? W[(long)bn * ldw + ka] : 0.f;
      b.y = (bn < N && ka + 1 < K) ? W[(long)bn * ldw + ka + 1] : 0.f;
    } else {
      b.x = (bn < N && ka < K) ? W[(long)ka * ldw + bn] : 0.f;
      b.y = (bn < N && ka + 1 < K) ? W[(long)(ka + 1) * ldw + bn] : 0.f;
    }
    acc = wmma4(a, b, acc);
  }
  const int col = tn + l16;
  if (col < N) {
    const float bv = bias ? bias[col] : 0.f;
#pragma unroll
    for (int r = 0; r < 8; ++r) {
      const int row = tm + half * 8 + r;
      if (row < M) C[(long)row * ldc + col] = acc[r] + bv;
    }
  }
}

// ---------------------------------------------------------------------------
// Fused attention: one workgroup (16 waves, 512 thr) owns FA_ROWS q-rows for
// one (b,h). Scores live entirely in LDS (exact full-row softmax, same
// semantics as reference). grid = (S/FA_ROWS, B*H).
//   Phase 1: QK^T via WMMA -> scaled/masked scores -> LDS
//   Phase 2: wave-per-row softmax in LDS
//   Phase 3: O = P@V, A-fragments from LDS (ds_load_b64), V from global
// ---------------------------------------------------------------------------
template <bool CAUSAL>
__global__ __launch_bounds__(512) void fattn_k(const float* __restrict__ q,
                                               const float* __restrict__ kk,
                                               const float* __restrict__ v,
                                               float* __restrict__ o) {
  extern __shared__ float P[];  // [FA_ROWS][FA_STRIDE]
  const int z = blockIdx.y;
  const int qb = blockIdx.x * FA_ROWS;
  const long base = (long)z * Sn * DHn;
  const float* Q = q + base;
  const float* Kp = kk + base;
  const float* V = v + base;
  const int wave = threadIdx.x >> 5;
  const int lane = threadIdx.x & 31;
  const int half = lane >> 4;
  const int l16 = lane & 15;
  const int kh = half * 2;
  const float scale = 0.125f;  // 1/sqrt(DH)

  // ---- Phase 1: scores. wave w -> key-column block w (32 cols) ----
  const int cb = wave * 32;
  const int c0 = cb + l16;
  const int c1 = cb + 16 + l16;
#pragma unroll
  for (int rblk = 0; rblk < FA_ROWS / 32; ++rblk) {
    const int rb0 = qb + rblk * 32;
    const int rbl = rblk * 32 + half * 8;  // local row base for C layout
    if (CAUSAL && cb > rb0 + 31) {         // tile fully masked (wave-uniform)
#pragma unroll
      for (int r = 0; r < 8; ++r) {
        P[(rbl + r) * FA_STRIDE + c0] = -1e9f;
        P[(rbl + r) * FA_STRIDE + c1] = -1e9f;
        P[(rbl + 16 + r) * FA_STRIDE + c0] = -1e9f;
        P[(rbl + 16 + r) * FA_STRIDE + c1] = -1e9f;
      }
      continue;
    }
    const float* Ar0 = Q + (long)(rb0 + l16) * DHn + kh;
    const float* Ar1 = Q + (long)(rb0 + 16 + l16) * DHn + kh;
    const float* Br0 = Kp + (long)(cb + l16) * DHn + kh;
    const float* Br1 = Kp + (long)(cb + 16 + l16) * DHn + kh;
    v8f a00 = {}, a01 = {}, a10 = {}, a11 = {};
#pragma unroll
    for (int k0 = 0; k0 < DHn; k0 += 4) {
      const v2f qa0 = *(const v2f*)(Ar0 + k0);
      const v2f qa1 = *(const v2f*)(Ar1 + k0);
      const v2f kb0 = *(const v2f*)(Br0 + k0);
      const v2f kb1 = *(const v2f*)(Br1 + k0);
      a00 = wmma4(qa0, kb0, a00);
      a01 = wmma4(qa0, kb1, a01);
      a10 = wmma4(qa1, kb0, a10);
      a11 = wmma4(qa1, kb1, a11);
    }
#pragma unroll
    for (int r = 0; r < 8; ++r) {
      const int lr0 = rbl + r;
      const int lr1 = rbl + 16 + r;
      const int gr0 = qb + lr0;
      const int gr1 = qb + lr1;
      float s00 = a00[r] * scale, s01 = a01[r] * scale;
      float s10 = a10[r] * scale, s11 = a11[r] * scale;
      if (CAUSAL) {
        if (c0 > gr0) s00 = -1e9f;
        if (c1 > gr0) s01 = -1e9f;
        if (c0 > gr1) s10 = -1e9f;
        if (c1 > gr1) s11 = -1e9f;
      }
      P[lr0 * FA_STRIDE + c0] = s00;
      P[lr0 * FA_STRIDE + c1] = s01;
      P[lr1 * FA_STRIDE + c0] = s10;
      P[lr1 * FA_STRIDE + c1] = s11;
    }
  }
  __syncthreads();

  // ---- Phase 2: softmax, wave per row (FA_ROWS/16 rows per wave) ----
#pragma unroll
  for (int t = 0; t < FA_ROWS / 16; ++t) {
    const int row = wave * (FA_ROWS / 16) + t;
    float* pr = P + row * FA_STRIDE;
    float vals[16];
    float mx = -3.4e38f;
#pragma unroll
    for (int i = 0; i < 16; ++i) {
      vals[i] = pr[lane + i * 32];
      mx = fmaxf(mx, vals[i]);
    }
#pragma unroll
    for (int os = 16; os > 0; os >>= 1) mx = fmaxf(mx, __shfl_xor(mx, os, 32));
    float sum = 0.f;
#pragma unroll
    for (int i = 0; i < 16; ++i) {
      vals[i] = expf(vals[i] - mx);
      sum += vals[i];
    }
#pragma unroll
    for (int os = 16; os > 0; os >>= 1) sum += __shfl_xor(sum, os, 32);
    const float rinv = 1.f / sum;
#pragma unroll
    for (int i = 0; i < 16; ++i) pr[lane + i * 32] = vals[i] * rinv;
  }
  __syncthreads();

  // ---- Phase 3: O = P @ V. 16 waves x one 16x16 tile (FA_ROWS/16 x 4) ----
  const int rt = (wave >> 2) * 16;  // local O row base
  const int ct = (wave & 3) * 16;   // O col base
  const float* Alds = P + (long)(rt + l16) * FA_STRIDE + kh;
  const float* Vc = V + ct + l16;
  v8f oc = {};
#pragma unroll 8
  for (int k0 = 0; k0 < Sn; k0 += 4) {
    const v2f a = *(const v2f*)(Alds + k0);
    v2f b;
    b.x = Vc[(long)(k0 + kh) * DHn];
    b.y = Vc[(long)(k0 + kh + 1) * DHn];
    oc = wmma4(a, b, oc);
  }
  float* O = o + base;
#pragma unroll
  for (int r = 0; r < 8; ++r)
    O[(long)(qb + rt + half * 8 + r) * DHn + ct + l16] = oc[r];
}

// ---------------------------------------------------------------------------
// LayerNorm over last dim (E=512): one wave per row, 16 elems per lane.
// ---------------------------------------------------------------------------
__global__ __launch_bounds__(128) void layernorm_k(
    const float* __restrict__ in, const float* __restrict__ sc,
    const float* __restrict__ bi, float* __restrict__ out, int rows) {
  const int wave = threadIdx.x >> 5;
  const int lane = threadIdx.x & 31;
  const int row = blockIdx.x * 4 + wave;
  if (row >= rows) return;
  const float* x = in + (long)row * En;
  float vals[16];
  float m = 0.f;
#pragma unroll
  for (int i = 0; i < 16; ++i) {
    vals[i] = x[lane + i * 32];
    m += vals[i];
  }
#pragma unroll
  for (int o = 16; o > 0; o >>= 1) m += __shfl_xor(m, o, 32);
  m *= (1.f / En);
  float v = 0.f;
#pragma unroll
  for (int i = 0; i < 16; ++i) {
    const float d = vals[i] - m;
    v += d * d;
  }
#pragma unroll
  for (int o = 16; o > 0; o >>= 1) v += __shfl_xor(v, o, 32);
  v *= (1.f / En);
  const float r = rsqrtf(v + 1e-5f);
  float* y = out + (long)row * En;
#pragma unroll
  for (int i = 0; i < 16; ++i) {
    const int c = lane + i * 32;
    y[c] = (vals[i] - m) * r * sc[c] + bi[c];
  }
}

// ---------------------------------------------------------------------------
// Head repacks (fused with RoPE where possible)
// ---------------------------------------------------------------------------
__global__ void split_qkv_rope_k(const float* __restrict__ qkv,
                                 float* __restrict__ q, float* __restrict__ k,
                                 float* __restrict__ v) {
  const size_t i = (size_t)blockIdx.x * blockDim.x + threadIdx.x;
  if (i >= (size_t)Bn * Hn * Sn * (DHn / 2)) return;
  const int j = i & (DHn / 2 - 1);
  const size_t bhs = i / (DHn / 2);
  const int s = bhs % Sn;
  const int hh = (bhs / Sn) % Hn;
  const int b = bhs / ((size_t)Sn * Hn);
  const size_t src = ((size_t)(b * Sn + s)) * (3 * En) + hh * DHn;
  const size_t o0 = bhs * DHn + j;
  const size_t o1 = o0 + DHn / 2;
  const float inv = powf(10000.f, -((float)(2 * j)) / (float)DHn);
  const float ang = (float)s * inv;
  const float c = cosf(ang);
  const float sn = sinf(ang);
  const float q1 = qkv[src + j], q2 = qkv[src + j + DHn / 2];
  const float k1 = qkv[src + En + j], k2 = qkv[src + En + j + DHn / 2];
  q[o0] = q1 * c - q2 * sn;
  q[o1] = q1 * sn + q2 * c;
  k[o0] = k1 * c - k2 * sn;
  k[o1] = k1 * sn + k2 * c;
  v[o0] = qkv[src + 2 * En + j];
  v[o1] = qkv[src + 2 * En + j + DHn / 2];
}

__global__ void split_heads_k(const float* __restrict__ in,
                              float* __restrict__ out, int rowW, int colOff) {
  const size_t i = (size_t)blockIdx.x * blockDim.x + threadIdx.x;
  if (i >= (size_t)Mrows * En) return;
  const int d = i % DHn;
  const int s = (i / DHn) % Sn;
  const int hh = (i / ((size_t)DHn * Sn)) % Hn;
  const int b = i / ((size_t)DHn * Sn * Hn);
  out[i] = in[((size_t)(b * Sn + s)) * rowW + colOff + hh * DHn + d];
}

__global__ void split_kv_k(const float* __restrict__ in, float* __restrict__ k,
                           float* __restrict__ v) {
  const size_t i = (size_t)blockIdx.x * blockDim.x + threadIdx.x;
  if (i >= (size_t)Mrows * En) return;
  const int d = i % DHn;
  const int s = (i / DHn) % Sn;
  const int hh = (i / ((size_t)DHn * Sn)) % Hn;
  const int b = i / ((size_t)DHn * Sn * Hn);
  const size_t src = ((size_t)(b * Sn + s)) * (2 * En) + hh * DHn + d;
  k[i] = in[src];
  v[i] = in[src + En];
}

__global__ void merge_heads_k(const float* __restrict__ in,
                              float* __restrict__ out) {
  const size_t j = (size_t)blockIdx.x * blockDim.x + threadIdx.x;
  if (j >= (size_t)Mrows * En) return;
  const int e = j % En;
  const int s = (j / En) % Sn;
  const int b = j / ((size_t)En * Sn);
  const int hh = e / DHn;
  const int d = e % DHn;
  out[j] = in[(((size_t)b * Hn + hh) * Sn + s) * DHn + d];
}

// ---------------------------------------------------------------------------
// Sigmoid gates -> leaf probabilities + KL accumulation
// ---------------------------------------------------------------------------
__global__ void gate_tree_k(const float* __restrict__ g, float* __restrict__ lp,
                            float* __restrict__ kl) {
  const int i = blockIdx.x * blockDim.x + threadIdx.x;
  if (i >= Mrows) return;
  float pr[NINn];
  float part = 0.f;
#pragma unroll
  for (int n = 0; n < NINn; ++n) {
    const float p = 1.f / (1.f + expf(-g[i * NINn + n]));
    pr[n] = p;
    const float q = fminf(fmaxf(p, 1e-6f), 1.f - 1e-6f);
    part += q * logf(2.f * q) + (1.f - q) * logf(2.f * (1.f - q));
  }
  atomicAdd(kl, part * (1.f / (float)(Mrows * NINn)));
#pragma unroll
  for (int leaf = 0; leaf < NLEAFn; ++leaf) {
    float prob = 1.f;
    int node = 0;
#pragma unroll
    for (int d = 0; d < 3; ++d) {
      const int bit = (leaf >> (2 - d)) & 1;
      prob *= bit ? pr[node] : (1.f - pr[node]);
      node = 2 * node + 1 + bit;
    }
    lp[i * NLEAFn + leaf] = prob;
  }
}

__global__ void leaf_accum_k(float* __restrict__ x, const float* __restrict__ lo,
                             const float* __restrict__ lp, int leaf) {
  const size_t j = (size_t)blockIdx.x * blockDim.x + threadIdx.x;
  if (j >= (size_t)Mrows * En) return;
  x[j] += lo[j] * lp[(j / En) * NLEAFn + leaf];
}

__global__ void leaf_sum_k(float* __restrict__ x, const float* __restrict__ lo,
                           const float* __restrict__ lp) {
  const size_t j = (size_t)blockIdx.x * blockDim.x + threadIdx.x;
  if (j >= (size_t)Mrows * En) return;
  const size_t row = j / En;
  float s = 0.f;
#pragma unroll
  for (int l = 0; l < NLEAFn; ++l)
    s += lo[(size_t)l * Mrows * En + j] * lp[row * NLEAFn + l];
  x[j] += s;
}

__global__ void zero_one_k(float* __restrict__ p) { p[0] = 0.f; }

// ---------------------------------------------------------------------------
// Host side
// ---------------------------------------------------------------------------
static inline int cdiv(int a, int b) { return (a + b - 1) / b; }

static void gemm(hipStream_t st, const float* A, int lda, long sA,
                 const float* W, int ldw, long sW, const float* bias,
                 long sBias, float* C, int ldc, long sC, int M, int N, int K,
                 int Z, bool transb, int act, bool accum) {
  if ((M % 128 == 0) && (N % 32 == 0) && (K % 4 == 0) && K >= 4) {
    dim3 g(N / 32, M / 128, Z), b(32, 4);
    if (transb)
      gemm_wmma_fast<true, 0, false><<<g, b, 0, st>>>(A, lda, sA, W, ldw, sW, bias, sBias, C, ldc, sC, K);
    else if (accum)
      gemm_wmma_fast<false, 0, true><<<g, b, 0, st>>>(A, lda, sA, W, ldw, sW, bias, sBias, C, ldc, sC, K);
    else if (act == 1)
      gemm_wmma_fast<false, 1, false><<<g, b, 0, st>>>(A, lda, sA, W, ldw, sW, bias, sBias, C, ldc, sC, K);
    else if (act == 2)
      gemm_wmma_fast<false, 2, false><<<g, b, 0, st>>>(A, lda, sA, W, ldw, sW, bias, sBias, C, ldc, sC, K);
    else
      gemm_wmma_fast<false, 0, false><<<g, b, 0, st>>>(A, lda, sA, W, ldw, sW, bias, sBias, C, ldc, sC, K);
  } else {
    dim3 g(cdiv(N, 16), cdiv(M, 64), Z), b(32, 4);
    if (transb)
      gemm_wmma_ref<true><<<g, b, 0, st>>>(A, lda, sA, W, ldw, sW, bias, C, ldc, sC, M, N, K);
    else
      gemm_wmma_ref<false><<<g, b, 0, st>>>(A, lda, sA, W, ldw, sW, bias, C, ldc, sC, M, N, K);
  }
}

extern "C" void kernel_launch(void* const* d_in, const int* in_sizes, int n_in,
                              void* d_out, int out_size, void* d_ws,
                              size_t ws_size, hipStream_t stream) {
  (void)in_sizes; (void)n_in; (void)out_size;
  const float* P[40];
  for (int i = 0; i < 37; ++i) P[i] = (const float*)d_in[i];
  const float* states = P[0];
  const float* actions = P[1];
  const float* rtg = P[2];
  const float *we_s = P[3], *be_s = P[4], *we_a = P[5], *be_a = P[6],
              *we_r = P[7], *be_r = P[8], *ln1_s = P[9], *ln1_b = P[10],
              *wqkv = P[11], *bqkv = P[12], *wo = P[13], *bo = P[14],
              *ln2_s = P[15], *ln2_b = P[16], *wq_c = P[17], *bq_c = P[18],
              *wkv_c = P[19], *bkv_c = P[20], *wo_c = P[21], *bo_c = P[22],
              *ln3_s = P[23], *ln3_b = P[24], *gate_w = P[25], *gate_b = P[26],
              *lw1 = P[27], *lb1 = P[28], *lw2 = P[29], *lb2 = P[30],
              *wp = P[31], *bp = P[32], *wr = P[33], *br = P[34], *wv = P[35],
              *bv = P[36];

  // Allow the fused-attention kernels to use 132 KB of dynamic LDS.
  hipFuncSetAttribute((const void*)fattn_k<true>,
                      hipFuncAttributeMaxDynamicSharedMemorySize, FA_LDS_BYTES);
  hipFuncSetAttribute((const void*)fattn_k<false>,
                      hipFuncAttributeMaxDynamicSharedMemorySize, FA_LDS_BYTES);

  const size_t BSE = (size_t)Mrows * En;
  const size_t baseFloats =
      11 * BSE + (size_t)Mrows * NINn + (size_t)Mrows * NLEAFn;
  const size_t avail = ws_size / sizeof(float);
  const bool bigLeaf = avail >= baseFloats + 16 * BSE + 64;

  float* w = (float*)d_ws;
  float* x = w;    w += BSE;
  float* ctx = w;  w += BSE;
  float* h = w;    w += BSE;
  float* buf3 = w; w += 3 * BSE;  // qkv / kv_c ; reused as mid/lo when !bigLeaf
  float* q_ = w;   w += BSE;
  float* k_ = w;   w += BSE;
  float* v_ = w;   w += BSE;
  float* o_ = w;   w += BSE;
  float* om = w;   w += BSE;
  float* gl = w;   w += (size_t)Mrows * NINn;
  float* lp = w;   w += (size_t)Mrows * NLEAFn;
  float* mid8 = bigLeaf ? w : buf3;
  if (bigLeaf) w += 8 * BSE;
  float* lo8 = bigLeaf ? w : buf3 + BSE;
  if (bigLeaf) w += 8 * BSE;

  float* out = (float*)d_out;
  float* out_act = out;
  float* out_state = out_act + (size_t)Mrows * ACTn;
  float* out_val = out_state + (size_t)Mrows * STATEn;
  float* out_kl = out_val + Mrows;

  const int EW = 256;
  const int EWG = (int)((BSE + EW - 1) / EW);
  const int HWG = (int)(((size_t)Bn * Hn * Sn * (DHn / 2) + EW - 1) / EW);
  const dim3 faGrid(Sn / FA_ROWS, Bn * Hn);

  zero_one_k<<<1, 1, 0, stream>>>(out_kl);

  // embeddings
  gemm(stream, states, STATEn, 0, we_s, En, 0, be_s, 0, x, En, 0, Mrows, En, STATEn, 1, false, 0, false);
  gemm(stream, actions, ACTn, 0, we_a, En, 0, be_a, 0, x, En, 0, Mrows, En, ACTn, 1, false, 0, true);
  gemm(stream, rtg, 1, 0, we_r, En, 0, be_r, 0, ctx, En, 0, Mrows, En, 1, 1, false, 0, false);

  for (int i = 0; i < Ln; ++i) {
    // ---- self-attention ----
    layernorm_k<<<cdiv(Mrows, 4), 128, 0, stream>>>(x, ln1_s + i * En, ln1_b + i * En, h, Mrows);
    gemm(stream, h, En, 0, wqkv + (size_t)i * En * 3 * En, 3 * En, 0,
         bqkv + (size_t)i * 3 * En, 0, buf3, 3 * En, 0, Mrows, 3 * En, En, 1, false, 0, false);
    split_qkv_rope_k<<<HWG, EW, 0, stream>>>(buf3, q_, k_, v_);
    fattn_k<true><<<faGrid, 512, FA_LDS_BYTES, stream>>>(q_, k_, v_, o_);
    merge_heads_k<<<EWG, EW, 0, stream>>>(o_, om);
    gemm(stream, om, En, 0, wo + (size_t)i * En * En, En, 0, bo + (size_t)i * En,
         0, x, En, 0, Mrows, En, En, 1, false, 0, true);

    // ---- cross-attention against ctx ----
    layernorm_k<<<cdiv(Mrows, 4), 128, 0, stream>>>(x, ln2_s + i * En, ln2_b + i * En, h, Mrows);
    gemm(stream, h, En, 0, wq_c + (size_t)i * En * En, En, 0, bq_c + (size_t)i * En,
         0, om, En, 0, Mrows, En, En, 1, false, 0, false);
    split_heads_k<<<EWG, EW, 0, stream>>>(om, q_, En, 0);
    gemm(stream, ctx, En, 0, wkv_c + (size_t)i * En * 2 * En, 2 * En, 0,
         bkv_c + (size_t)i * 2 * En, 0, buf3, 2 * En, 0, Mrows, 2 * En, En, 1, false, 0, false);
    split_kv_k<<<EWG, EW, 0, stream>>>(buf3, k_, v_);
    fattn_k<false><<<faGrid, 512, FA_LDS_BYTES, stream>>>(q_, k_, v_, o_);
    merge_heads_k<<<EWG, EW, 0, stream>>>(o_, om);
    gemm(stream, om, En, 0, wo_c + (size_t)i * En * En, En, 0,
         bo_c + (size_t)i * En, 0, x, En, 0, Mrows, En, En, 1, false, 0, true);

    // ---- soft decision-tree MoE ----
    layernorm_k<<<cdiv(Mrows, 4), 128, 0, stream>>>(x, ln3_s + i * En, ln3_b + i * En, h, Mrows);
    gemm(stream, h, En, 0, gate_w + (size_t)i * NINn * En, En, 0,
         gate_b + (size_t)i * NINn, 0, gl, NINn, 0, Mrows, NINn, En, 1, true, 0, false);
    gate_tree_k<<<cdiv(Mrows, 256), 256, 0, stream>>>(gl, lp, out_kl);
    const float* lw1_i = lw1 + (size_t)i * NLEAFn * En * FHn;
    const float* lb1_i = lb1 + (size_t)i * NLEAFn * FHn;
    const float* lw2_i = lw2 + (size_t)i * NLEAFn * FHn * En;
    const float* lb2_i = lb2 + (size_t)i * NLEAFn * En;
    if (bigLeaf) {
      gemm(stream, h, En, 0, lw1_i, FHn, (long)En * FHn, lb1_i, FHn, mid8, FHn,
           (long)Mrows * FHn, Mrows, FHn, En, NLEAFn, false, 1, false);
      gemm(stream, mid8, FHn, (long)Mrows * FHn, lw2_i, En, (long)FHn * En,
           lb2_i, En, lo8, En, (long)Mrows * En, Mrows, En, FHn, NLEAFn, false, 0, false);
      leaf_sum_k<<<EWG, EW, 0, stream>>>(x, lo8, lp);
    } else {
      for (int leaf = 0; leaf < NLEAFn; ++leaf) {
        gemm(stream, h, En, 0, lw1_i + (size_t)leaf * En * FHn, FHn, 0,
             lb1_i + (size_t)leaf * FHn, 0, mid8, FHn, 0, Mrows, FHn, En, 1, false, 1, false);
        gemm(stream, mid8, FHn, 0, lw2_i + (size_t)leaf * FHn * En, En, 0,
             lb2_i + (size_t)leaf * En, 0, lo8, En, 0, Mrows, En, FHn, 1, false, 0, false);
        leaf_accum_k<<<EWG, EW, 0, stream>>>(x, lo8, lp, leaf);
      }
    }
  }

  // ---- output heads ----
  gemm(stream, x, En, 0, wp, ACTn, 0, bp, 0, out_act, ACTn, 0, Mrows, ACTn, En, 1, false, 2, false);
  gemm(stream, x, En, 0, wr, STATEn, 0, br, 0, out_state, STATEn, 0, Mrows, STATEn, En, 1, false, 0, false);
  gemm(stream, x, En, 0, wv, 1, 0, bv, 0, out_val, 1, 0, Mrows, 1, En, 1, false, 0, false);
}